// S_LSTM_39410619908256
// MI455X (gfx1250) — compile-verified
//
#include <hip/hip_runtime.h>

// Problem constants (match reference)
#define TT 16
#define HH 1024
#define NL 4

#ifndef __has_builtin
#define __has_builtin(x) 0
#endif

// gfx1250 async global->LDS copy path (guarded; falls back to plain LDS store)
#if __has_builtin(__builtin_amdgcn_global_load_async_to_lds_b32) && \
    __has_builtin(__builtin_amdgcn_s_wait_asynccnt)
#define USE_ASYNC_LDS 1
#else
#define USE_ASYNC_LDS 0
#endif

typedef __attribute__((address_space(1))) int global_i32;
typedef __attribute__((address_space(3))) int lds_i32;

// gfx1250 v_tanh_f32 (guarded; falls back to exp/rcp formulation)
#if __has_builtin(__builtin_amdgcn_tanhf)
__device__ __forceinline__ float t_tanh(float x) { return __builtin_amdgcn_tanhf(x); }
#else
__device__ __forceinline__ float t_tanh(float x) {
  // tanh(x) = 1 - 2/(exp(2x)+1); saturates correctly at +/-1 for large |x|
  float e = __expf(2.0f * x);
  return 1.0f - 2.0f * __builtin_amdgcn_rcpf(e + 1.0f);
}
#endif

__device__ __forceinline__ float t_sigm(float x) {
  // sigmoid(x) = 0.5 * (1 + tanh(x/2))
  return fmaf(0.5f, t_tanh(0.5f * x), 0.5f);
}

// One thread per element (r,c) of the HxH state. All 64 (t,layer) steps of
// that element's chain run in registers; only weights are read (once) and the
// (T,2,NL,H,H) output is written (once).
__global__ __launch_bounds__(256) void lstm_elem_kernel(
    const float* __restrict__ x,
    const float* __restrict__ Wxi, const float* __restrict__ Wxf,
    const float* __restrict__ Wxc, const float* __restrict__ Wxo,
    const float* __restrict__ Whi, const float* __restrict__ Whf,
    const float* __restrict__ Whc, const float* __restrict__ Who,
    const float* __restrict__ Whsi, const float* __restrict__ Whsf,
    const float* __restrict__ Whsc, const float* __restrict__ Whso,
    const float* __restrict__ bi, const float* __restrict__ bf,
    const float* __restrict__ bc, const float* __restrict__ bo,
    float* __restrict__ out)
{
  const int idx = blockIdx.x * blockDim.x + threadIdx.x;  // 0 .. H*H-1
  const int r = idx >> 10;        // row    (H = 1024)
  const int c = idx & (HH - 1);   // column

  // Each 256-thread block covers a single row r (1024 % 256 == 0), so the
  // 16 x-values x[t][r] are shared by the whole block: broadcast via LDS
  // using the gfx1250 async copy engine (one b32 per lane 0..15 of wave 0).
  __shared__ float xs[TT];
  if (threadIdx.x < TT) {
    const float* gp = x + threadIdx.x * HH + r;   // x layout (T, H, 1)
#if USE_ASYNC_LDS
    __builtin_amdgcn_global_load_async_to_lds_b32(
        (global_i32*)gp,
        (lds_i32*)&xs[threadIdx.x],
        /*imm offset*/0, /*cpol*/0);
#else
    xs[threadIdx.x] = *gp;
#endif
  }

  // ---- Load all per-element parameters into registers (read exactly once;
  // reused across all 16 timesteps -> minimal HBM traffic: 112 MB total). ----
  float whi[NL], whf[NL], whc[NL], who[NL];     // recurrent (H,H) per layer
  float wxi[NL], wxf[NL], wxc[NL], wxo[NL];     // input->gate (1,H) per layer
  float vbi[NL], vbf[NL], vbc[NL], vbo[NL];     // biases (1,H) per layer
  float wsi[NL], wsf[NL], wsc[NL], wso[NL];     // layer-skip (H,H), l>=1

#pragma unroll
  for (int l = 0; l < NL; ++l) {
    const int eo = l * (HH * HH) + idx;
    whi[l] = Whi[eo]; whf[l] = Whf[eo]; whc[l] = Whc[eo]; who[l] = Who[eo];
    const int co = l * HH + c;
    wxi[l] = Wxi[co]; wxf[l] = Wxf[co]; wxc[l] = Wxc[co]; wxo[l] = Wxo[co];
    vbi[l] = bi[co];  vbf[l] = bf[co];  vbc[l] = bc[co];  vbo[l] = bo[co];
  }
  wsi[0] = wsf[0] = wsc[0] = wso[0] = 0.0f;
#pragma unroll
  for (int l = 1; l < NL; ++l) {
    const int so = (l - 1) * (HH * HH) + idx;
    wsi[l] = Whsi[so]; wsf[l] = Whsf[so]; wsc[l] = Whsc[so]; wso[l] = Whso[so];
  }

#if USE_ASYNC_LDS
  __builtin_amdgcn_s_wait_asynccnt(0);   // drain async copies (issuing wave)
#endif
  __syncthreads();

  // ---- Recurrence fully in registers ----
  float cp[NL] = {0.f, 0.f, 0.f, 0.f};   // c at t-1
  float hp[NL] = {0.f, 0.f, 0.f, 0.f};   // h at t-1

  float* outp = out + idx;               // out layout (T, 2, NL, H, H)

  for (int t = 0; t < TT; ++t) {
    const float xt = xs[t];
    float cc[NL], hc[NL];

#pragma unroll
    for (int l = 0; l < NL; ++l) {
      float zi = fmaf(xt, wxi[l], vbi[l]);
      float zf = fmaf(xt, wxf[l], vbf[l]);
      float zc = fmaf(xt, wxc[l], vbc[l]);
      float zo = fmaf(xt, wxo[l], vbo[l]);
      zi = fmaf(hp[l], whi[l], zi);
      zf = fmaf(hp[l], whf[l], zf);
      zc = fmaf(hp[l], whc[l], zc);
      zo = fmaf(hp[l], who[l], zo);
      if (l > 0) {                       // layer-skip uses CURRENT-t h below
        const float hb = hc[l - 1];
        zi = fmaf(hb, wsi[l], zi);
        zf = fmaf(hb, wsf[l], zf);
        zc = fmaf(hb, wsc[l], zc);
        zo = fmaf(hb, wso[l], zo);
      }
      const float gi = t_sigm(zi);
      const float gf = t_sigm(zf);
      const float go = t_sigm(zo);
      const float gc = t_tanh(zc);
      const float cn = fmaf(gf, cp[l], gi * gc);
      cc[l] = cn;
      hc[l] = go * t_tanh(cn);
    }

    // Streaming write-once output: nontemporal stores (512 MB total).
#pragma unroll
    for (int l = 0; l < NL; ++l) {
      __builtin_nontemporal_store(cc[l], outp + ((size_t)((t * 2)     * NL + l) << 20));
      __builtin_nontemporal_store(hc[l], outp + ((size_t)((t * 2 + 1) * NL + l) << 20));
      cp[l] = cc[l];
      hp[l] = hc[l];
    }
  }
}

extern "C" void kernel_launch(void* const* d_in, const int* in_sizes, int n_in,
                              void* d_out, int out_size, void* d_ws, size_t ws_size,
                              hipStream_t stream) {
  (void)in_sizes; (void)n_in; (void)out_size; (void)d_ws; (void)ws_size;

  const float* x    = (const float*)d_in[0];
  const float* Wxi  = (const float*)d_in[1];
  const float* Wxf  = (const float*)d_in[2];
  const float* Wxc  = (const float*)d_in[3];
  const float* Wxo  = (const float*)d_in[4];
  const float* Whi  = (const float*)d_in[5];
  const float* Whf  = (const float*)d_in[6];
  const float* Whc  = (const float*)d_in[7];
  const float* Who  = (const float*)d_in[8];
  const float* Whsi = (const float*)d_in[9];
  const float* Whsf = (const float*)d_in[10];
  const float* Whsc = (const float*)d_in[11];
  const float* Whso = (const float*)d_in[12];
  const float* bi   = (const float*)d_in[13];
  const float* bf   = (const float*)d_in[14];
  const float* bc   = (const float*)d_in[15];
  const float* bo   = (const float*)d_in[16];
  float* out = (float*)d_out;

  const int threads = 256;
  const int blocks  = (HH * HH) / threads;  // 4096 blocks, one row per block segment
  lstm_elem_kernel<<<blocks, threads, 0, stream>>>(
      x, Wxi, Wxf, Wxc, Wxo, Whi, Whf, Whc, Who,
      Whsi, Whsf, Whsc, Whso, bi, bf, bc, bo, out);
}